// VariationalGraohAutoDecoder_9045201126084
// MI455X (gfx1250) — compile-verified
//
#include <hip/hip_runtime.h>
#include <hip/hip_bf16.h>

typedef __attribute__((ext_vector_type(2))) float v2f;
typedef __attribute__((ext_vector_type(8))) float v8f;

#define D_HID 128
#define D_OUT 64

// ---------------------------------------------------------------------------
// Utility: zero a float buffer (grid-strided)
// ---------------------------------------------------------------------------
__global__ void zero_f32(float* __restrict__ p, long n) {
    long i = (long)blockIdx.x * blockDim.x + threadIdx.x;
    long stride = (long)gridDim.x * blockDim.x;
    for (; i < n; i += stride) p[i] = 0.0f;
}

// ---------------------------------------------------------------------------
// Per-edge in-degree count (f32 so it doubles as GCN degree base)
// ---------------------------------------------------------------------------
__global__ void count_deg(const int* __restrict__ dst, float* __restrict__ cnt, int E) {
    int i = blockIdx.x * blockDim.x + threadIdx.x;
    if (i < E) atomicAdd(&cnt[dst[i]], 1.0f);
}

// ---------------------------------------------------------------------------
// Scatter-add: msum[dst] += x[src], D=128. One thread = (edge, 4 dims).
// float4 loads; 4 f32 atomics (land in L2; 51MB dest fits in 192MB L2).
// ---------------------------------------------------------------------------
__global__ void scatter_add_128(const float* __restrict__ x,
                                const int* __restrict__ src,
                                const int* __restrict__ dst,
                                float* __restrict__ msum, int E) {
    long tid = (long)blockIdx.x * blockDim.x + threadIdx.x;
    long e = tid >> 5;
    if (e >= E) return;
    int d = (int)(tid & 31) * 4;
    long s = src[e], t = dst[e];
    float4 v = *(const float4*)(x + s * D_HID + d);
    float* o = msum + t * D_HID + d;
    atomicAdd(o + 0, v.x);
    atomicAdd(o + 1, v.y);
    atomicAdd(o + 2, v.z);
    atomicAdd(o + 3, v.w);
}

// msum -> mean  (divide by max(cnt,1))
__global__ void mean_div_128(float* __restrict__ msum, const float* __restrict__ cnt, long total) {
    long i = (long)blockIdx.x * blockDim.x + threadIdx.x;
    if (i >= total) return;
    float c = cnt[i >> 7];
    msum[i] *= 1.0f / fmaxf(c, 1.0f);
}

// ---------------------------------------------------------------------------
// FP32 WMMA GEMM: Out = A0@W0 [+ A1@W1 + bias] [ReLU]
// One wave -> one 16x16 output tile. K=KDIM in steps of 4 via
// V_WMMA_F32_16X16X4_F32 (wave32 matrix core, fp32-exact).
//
// VGPR layouts per CDNA5 ISA 7.12.2:
//  A (16x4 f32):  lanes 0-15 -> rows M=lane, {K=0,K=1}; lanes 16-31 -> {K=2,K=3}
//  B (4x16 f32):  lanes 0-15 -> cols N=lane, {K=0,K=1}; lanes 16-31 -> {K=2,K=3}
//  C (16x16 f32): VGPR r: lanes 0-15 = (M=r, N=lane); lanes 16-31 = (M=r+8, N=lane-16)
// ---------------------------------------------------------------------------
template <int KDIM, int NDIM, bool RELU, bool DUAL>
__global__ void gemm_wmma_f32(const float* __restrict__ A0,
                              const float* __restrict__ A1,
                              const float* __restrict__ W0,
                              const float* __restrict__ W1,
                              const float* __restrict__ bias,
                              float* __restrict__ Out, int nrows) {
    const int ntn = NDIM / 16;
    int wave = threadIdx.x >> 5;
    int lane = threadIdx.x & 31;
    long tile = (long)blockIdx.x * (blockDim.x >> 5) + wave;
    long m0 = (tile / ntn) * 16;
    int n0 = (int)(tile % ntn) * 16;
    if (m0 >= nrows) return;                 // uniform per wave: EXEC stays all-1s

    const int l = lane & 15;
    const int hi = lane >> 4;                // selects K pair {0,1} vs {2,3}

    v8f c = {};

    {
        const float* arow = A0 + (m0 + l) * (long)KDIM;
#pragma unroll 8
        for (int kk = 0; kk < KDIM; kk += 4) {
            v2f a = *(const v2f*)(arow + kk + 2 * hi);
            v2f b;
            b.x = W0[(long)(kk + 2 * hi) * NDIM + n0 + l];
            b.y = W0[(long)(kk + 2 * hi + 1) * NDIM + n0 + l];
            c = __builtin_amdgcn_wmma_f32_16x16x4_f32(false, a, false, b,
                                                      (short)0, c, false, false);
        }
    }
    if (DUAL) {
        const float* arow = A1 + (m0 + l) * (long)KDIM;
#pragma unroll 8
        for (int kk = 0; kk < KDIM; kk += 4) {
            v2f a = *(const v2f*)(arow + kk + 2 * hi);
            v2f b;
            b.x = W1[(long)(kk + 2 * hi) * NDIM + n0 + l];
            b.y = W1[(long)(kk + 2 * hi + 1) * NDIM + n0 + l];
            c = __builtin_amdgcn_wmma_f32_16x16x4_f32(false, a, false, b,
                                                      (short)0, c, false, false);
        }
    }

    float bv = DUAL ? bias[n0 + l] : 0.0f;
#pragma unroll
    for (int r = 0; r < 8; ++r) {
        long row = m0 + r + hi * 8;
        float v = c[r] + bv;
        if (RELU) v = fmaxf(v, 0.0f);
        Out[row * NDIM + n0 + l] = v;
    }
}

// ---------------------------------------------------------------------------
// GCN normalized scatter: agg[dst] += xw[src] * rsqrt(deg_s)*rsqrt(deg_t), D=64
// One thread = (edge, 4 dims). deg = cnt + 1 (self loop).
// ---------------------------------------------------------------------------
__global__ void gcn_scatter_64(const float* __restrict__ xw,
                               const int* __restrict__ src,
                               const int* __restrict__ dst,
                               const float* __restrict__ cnt,
                               float* __restrict__ agg, int E) {
    long tid = (long)blockIdx.x * blockDim.x + threadIdx.x;
    long e = tid >> 4;
    if (e >= E) return;
    int d = (int)(tid & 15) * 4;
    long s = src[e], t = dst[e];
    float norm = __frsqrt_rn(cnt[s] + 1.0f) * __frsqrt_rn(cnt[t] + 1.0f);
    float4 v = *(const float4*)(xw + s * D_OUT + d);
    float* o = agg + t * D_OUT + d;
    atomicAdd(o + 0, v.x * norm);
    atomicAdd(o + 1, v.y * norm);
    atomicAdd(o + 2, v.z * norm);
    atomicAdd(o + 3, v.w * norm);
}

// out = agg + xw/deg + bg;  sigmoid on cols [0,8)
__global__ void gcn_final_64(const float* __restrict__ agg,
                             const float* __restrict__ xw,
                             const float* __restrict__ cnt,
                             const float* __restrict__ bg,
                             float* __restrict__ out, long total) {
    long i = (long)blockIdx.x * blockDim.x + threadIdx.x;
    if (i >= total) return;
    long row = i >> 6;
    int col = (int)(i & 63);
    float deg = cnt[row] + 1.0f;
    float v = agg[i] + xw[i] / deg + bg[col];
    if (col < 8) v = 1.0f / (1.0f + __expf(-v));
    out[i] = v;
}

// ---------------------------------------------------------------------------
// Launch sequence
// ---------------------------------------------------------------------------
extern "C" void kernel_launch(void* const* d_in, const int* in_sizes, int n_in,
                              void* d_out, int out_size, void* d_ws, size_t ws_size,
                              hipStream_t stream) {
    const float* z   = (const float*)d_in[0];
    const int*   ei  = (const int*)d_in[1];
    const float* W1l = (const float*)d_in[2];
    const float* b1  = (const float*)d_in[3];
    const float* W1r = (const float*)d_in[4];
    const float* W2l = (const float*)d_in[5];
    const float* b2  = (const float*)d_in[6];
    const float* W2r = (const float*)d_in[7];
    const float* Wg  = (const float*)d_in[8];
    const float* bg  = (const float*)d_in[9];
    float* out = (float*)d_out;

    const int N = in_sizes[0] / D_HID;
    const int E = in_sizes[1] / 2;
    const int* src = ei;
    const int* dst = ei + E;

    // Workspace layout (aliased to stay compact):
    //   cnt  : N floats
    //   msum : N*128   (later reused as agg, N*64)
    //   x1   : N*128   (later reused as xw,  N*64)
    //   x2   : N*128
    float* ws   = (float*)d_ws;
    float* cnt  = ws;
    float* msum = cnt + N;
    float* x1   = msum + (size_t)N * D_HID;
    float* x2   = x1 + (size_t)N * D_HID;
    float* xw   = x1;    // alias (x1 dead after SAGE2)
    float* agg  = msum;  // alias (msum dead after SAGE2)

    const int TB = 256;
    auto cdiv = [](long a, long b) { return (unsigned)((a + b - 1) / b); };

    long nh = (long)N * D_HID;          // N*128
    long no = (long)N * D_OUT;          // N*64
    unsigned zb = 2048;                 // grid-strided zero blocks

    // degree counts
    zero_f32<<<zb, TB, 0, stream>>>(cnt, N);
    count_deg<<<cdiv(E, TB), TB, 0, stream>>>(dst, cnt, E);

    // ---- SAGE layer 1 ----
    zero_f32<<<zb, TB, 0, stream>>>(msum, nh);
    scatter_add_128<<<cdiv((long)E * 32, TB), TB, 0, stream>>>(z, src, dst, msum, E);
    mean_div_128<<<cdiv(nh, TB), TB, 0, stream>>>(msum, cnt, nh);
    {
        long tiles = ((long)(N + 15) / 16) * (D_HID / 16);
        gemm_wmma_f32<D_HID, D_HID, true, true>
            <<<cdiv(tiles, TB / 32), TB, 0, stream>>>(msum, z, W1l, W1r, b1, x1, N);
    }

    // ---- SAGE layer 2 ----
    zero_f32<<<zb, TB, 0, stream>>>(msum, nh);
    scatter_add_128<<<cdiv((long)E * 32, TB), TB, 0, stream>>>(x1, src, dst, msum, E);
    mean_div_128<<<cdiv(nh, TB), TB, 0, stream>>>(msum, cnt, nh);
    {
        long tiles = ((long)(N + 15) / 16) * (D_HID / 16);
        gemm_wmma_f32<D_HID, D_HID, true, true>
            <<<cdiv(tiles, TB / 32), TB, 0, stream>>>(msum, x1, W2l, W2r, b2, x2, N);
    }

    // ---- GCN layer ----  (x2 @ Wg -> xw; xw aliases x1, which is now dead)
    {
        long tiles = ((long)(N + 15) / 16) * (D_OUT / 16);
        gemm_wmma_f32<D_HID, D_OUT, false, false>
            <<<cdiv(tiles, TB / 32), TB, 0, stream>>>(x2, nullptr, Wg, nullptr, nullptr, xw, N);
    }
    zero_f32<<<zb, TB, 0, stream>>>(agg, no);
    gcn_scatter_64<<<cdiv((long)E * 16, TB), TB, 0, stream>>>(xw, src, dst, cnt, agg, E);
    gcn_final_64<<<cdiv(no, TB), TB, 0, stream>>>(agg, xw, cnt, bg, out, no);
}